// _GraphEncoder_42365557408073
// MI455X (gfx1250) — compile-verified
//
#include <hip/hip_runtime.h>
#include <hip/hip_bf16.h>

typedef __attribute__((ext_vector_type(2))) float v2f;
typedef __attribute__((ext_vector_type(8))) float v8f;

#define IN_CH     128
#define OUT_CH    128   // HEADS * HEAD_DIM
#define HEADS     4
#define HEAD_DIM  32
#define NEG_SLOPE 0.2f
#define LDS_STRIDE 132  // 128 + 4 pad -> conflict-free ds_load_b64 A fetches

// ---------------------------------------------------------------------------
// Kernel 1: xw = x @ W  via V_WMMA_F32_16X16X4_F32 (exact f32 precision).
// Workgroup = 8 waves = one 16-row x 128-col output block.
// Stage the 16x128 A block into LDS once (bounds-checked + zero-filled there),
// so the WMMA K-loop is branch-free: ds_load_b64 (A) + 2x global_load_b32 (B)
// + v_wmma per step. A-row garbage only reaches D rows >= N, whose stores are
// guarded, so no per-iteration masking is required (EXEC stays all-1s).
// ---------------------------------------------------------------------------
__global__ void gat_gemm_wmma(const float* __restrict__ x,
                              const float* __restrict__ W,
                              float* __restrict__ xw, int N) {
  __shared__ float axs[16 * LDS_STRIDE];

  const int tid  = threadIdx.x;
  const int lane = tid & 31;
  const int wave = tid >> 5;                  // 0..7 -> 16-col tile
  const int row0 = blockIdx.x * 16;

  // --- cooperative stage of A block: 2048 floats, 8 per thread (2x float4)
  {
    const int idx = tid * 8;                  // [0, 2048)
    const int r   = idx >> 7;                 // row within block
    const int c   = idx & 127;                // col
    const int row = row0 + r;
    float4 v0, v1;
    if (row < N) {
      const float4* p = (const float4*)(x + (size_t)row * IN_CH + c);
      v0 = p[0];
      v1 = p[1];
    } else {
      v0 = make_float4(0.f, 0.f, 0.f, 0.f);
      v1 = v0;
    }
    // (r*132 + c)*4 bytes is 16B aligned: 132 floats = 528 B (mult of 16),
    // c is a multiple of 8.
    float4* d = (float4*)(axs + r * LDS_STRIDE + c);
    d[0] = v0;
    d[1] = v1;
  }
  __syncthreads();

  const int g    = lane >> 4;                 // half-wave: 0 or 1
  const int lr   = lane & 15;
  const int col0 = wave * 16;

  v8f acc = {};
  for (int k = 0; k < IN_CH; k += 4) {
    // A: 16x4 f32 operand. lane l -> row l&15, K = k + 2*(l>>4) + {0,1}
    v2f a;
    const float* ap = axs + lr * LDS_STRIDE + k + 2 * g;
    a.x = ap[0];
    a.y = ap[1];
    // B: 4x16 f32 operand. lane l -> col col0+(l&15), rows k+2g, k+2g+1
    v2f b;
    b.x = W[(size_t)(k + 2 * g)     * OUT_CH + col0 + lr];
    b.y = W[(size_t)(k + 2 * g + 1) * OUT_CH + col0 + lr];
    acc = __builtin_amdgcn_wmma_f32_16x16x4_f32(
        /*neg_a=*/false, a, /*neg_b=*/false, b,
        /*c_mod=*/(short)0, acc, /*reuse_a=*/false, /*reuse_b=*/false);
  }

#pragma unroll
  for (int r = 0; r < 8; ++r) {
    int orow = row0 + r + 8 * g;              // C/D: VGPR r -> row r + 8*g
    if (orow < N) xw[(size_t)orow * OUT_CH + col0 + lr] = acc[r];
  }
}

// ---------------------------------------------------------------------------
// Kernel 2: per (node, head) attention logits + init of max/denom buffers.
// ---------------------------------------------------------------------------
__global__ void gat_node_att(const float* __restrict__ xw,
                             const float* __restrict__ att_src,
                             const float* __restrict__ att_dst,
                             float* __restrict__ a_src,
                             float* __restrict__ a_dst,
                             float* __restrict__ mx,
                             float* __restrict__ denom, int N) {
  int t = blockIdx.x * blockDim.x + threadIdx.x;
  if (t >= N * HEADS) return;
  int n = t >> 2, h = t & 3;
  const float* v  = xw + (size_t)n * OUT_CH + h * HEAD_DIM;
  const float* as = att_src + h * HEAD_DIM;
  const float* ad = att_dst + h * HEAD_DIM;
  float s1 = 0.0f, s2 = 0.0f;
#pragma unroll
  for (int i = 0; i < HEAD_DIM; ++i) {
    float vv = v[i];
    s1 += vv * as[i];
    s2 += vv * ad[i];
  }
  a_src[t] = s1;
  a_dst[t] = s2;
  mx[t]    = -__builtin_inff();
  denom[t] = 0.0f;
}

// out = bias (broadcast), so edge atomics accumulate on top.
__global__ void gat_init_out(float* __restrict__ out,
                             const float* __restrict__ bias, long long total) {
  long long i = blockIdx.x * (long long)blockDim.x + threadIdx.x;
  if (i < total) out[i] = bias[i & (OUT_CH - 1)];
}

__device__ __forceinline__ float leaky(float v) {
  return v > 0.0f ? v : NEG_SLOPE * v;
}

// float atomic max via signed/unsigned integer ordering trick
__device__ __forceinline__ void atomicMaxF(float* addr, float val) {
  if (val >= 0.0f)
    atomicMax((int*)addr, __float_as_int(val));
  else
    atomicMin((unsigned int*)addr, __float_as_uint(val));
}

__device__ __forceinline__ void edge_endpoints(const long long* __restrict__ ei,
                                               long long e, int E,
                                               int& src, int& dst) {
  if (e < (long long)E) {
    src = (int)ei[e];
    dst = (int)ei[(long long)E + e];
  } else {                 // PyG-default self loops appended after real edges
    src = dst = (int)(e - E);
  }
}

// ---------------------------------------------------------------------------
// Kernel 3: segment max of leaky-relu logits over incoming edges.
// ---------------------------------------------------------------------------
__global__ void gat_edge_max(const long long* __restrict__ ei,
                             const float* __restrict__ a_src,
                             const float* __restrict__ a_dst,
                             float* __restrict__ mx, int E, int N) {
  long long t = blockIdx.x * (long long)blockDim.x + threadIdx.x;
  long long total = (long long)(E + N) * HEADS;
  if (t >= total) return;
  int h = (int)(t & 3);
  long long e = t >> 2;
  int src, dst;
  edge_endpoints(ei, e, E, src, dst);
  float val = leaky(a_src[src * HEADS + h] + a_dst[dst * HEADS + h]);
  atomicMaxF(&mx[dst * HEADS + h], val);
}

// ---------------------------------------------------------------------------
// Kernel 4: segment sum of exp(logit - max).
// ---------------------------------------------------------------------------
__global__ void gat_edge_sum(const long long* __restrict__ ei,
                             const float* __restrict__ a_src,
                             const float* __restrict__ a_dst,
                             const float* __restrict__ mx,
                             float* __restrict__ denom, int E, int N) {
  long long t = blockIdx.x * (long long)blockDim.x + threadIdx.x;
  long long total = (long long)(E + N) * HEADS;
  if (t >= total) return;
  int h = (int)(t & 3);
  long long e = t >> 2;
  int src, dst;
  edge_endpoints(ei, e, E, src, dst);
  float val = leaky(a_src[src * HEADS + h] + a_dst[dst * HEADS + h]);
  float ex  = expf(val - mx[dst * HEADS + h]);
  atomicAdd(&denom[dst * HEADS + h], ex);
}

// ---------------------------------------------------------------------------
// Kernel 5: wave-per-edge weighted aggregation.
// 32 lanes x float4 = 128 channels, coalesced gather of xw[src] (L2-resident)
// and coalesced global_atomic_add_f32 scatter into out[dst].
// lane -> head: channels [4*lane, 4*lane+4) all live in head lane>>3.
// ---------------------------------------------------------------------------
__global__ void gat_edge_aggregate(const long long* __restrict__ ei,
                                   const float* __restrict__ a_src,
                                   const float* __restrict__ a_dst,
                                   const float* __restrict__ mx,
                                   const float* __restrict__ denom,
                                   const float4* __restrict__ xw4,
                                   float* __restrict__ out, int E, int N) {
  const int lane = threadIdx.x & 31;
  long long e = blockIdx.x * 8LL + (threadIdx.x >> 5);
  if (e >= (long long)E + N) return;
  int src, dst;
  edge_endpoints(ei, e, E, src, dst);
  const int h = lane >> 3;
  float val   = leaky(a_src[src * HEADS + h] + a_dst[dst * HEADS + h]);
  float alpha = expf(val - mx[dst * HEADS + h]) / denom[dst * HEADS + h];
  float4 v = xw4[(size_t)src * (OUT_CH / 4) + lane];
  float* op = out + (size_t)dst * OUT_CH + lane * 4;
  atomicAdd(op + 0, v.x * alpha);
  atomicAdd(op + 1, v.y * alpha);
  atomicAdd(op + 2, v.z * alpha);
  atomicAdd(op + 3, v.w * alpha);
}

// ---------------------------------------------------------------------------
extern "C" void kernel_launch(void* const* d_in, const int* in_sizes, int n_in,
                              void* d_out, int out_size, void* d_ws, size_t ws_size,
                              hipStream_t stream) {
  const float*     x       = (const float*)d_in[0];
  const float*     W       = (const float*)d_in[1];
  const float*     att_src = (const float*)d_in[2];
  const float*     att_dst = (const float*)d_in[3];
  const float*     bias    = (const float*)d_in[4];
  const long long* ei      = (const long long*)d_in[5];

  const int N = in_sizes[0] / IN_CH;
  const int E = in_sizes[5] / 2;

  float* out   = (float*)d_out;
  float* xw    = (float*)d_ws;                       // N*128 f32  (51.2 MB)
  float* a_src = xw    + (size_t)N * OUT_CH;         // N*4
  float* a_dst = a_src + (size_t)N * HEADS;          // N*4
  float* mx    = a_dst + (size_t)N * HEADS;          // N*4
  float* denom = mx    + (size_t)N * HEADS;          // N*4

  // 1) feature transform (WMMA f32, LDS-staged A)
  gat_gemm_wmma<<<(N + 15) / 16, 256, 0, stream>>>(x, W, xw, N);

  // 2) per-node logits + buffer init
  int nh = N * HEADS;
  gat_node_att<<<(nh + 255) / 256, 256, 0, stream>>>(xw, att_src, att_dst,
                                                     a_src, a_dst, mx, denom, N);
  long long total_out = (long long)N * OUT_CH;
  gat_init_out<<<(unsigned)((total_out + 255) / 256), 256, 0, stream>>>(out, bias, total_out);

  // 3) edge softmax (max pass, sum pass) over E real edges + N self loops
  long long eh = (long long)(E + N) * HEADS;
  unsigned ebl = (unsigned)((eh + 255) / 256);
  gat_edge_max<<<ebl, 256, 0, stream>>>(ei, a_src, a_dst, mx, E, N);
  gat_edge_sum<<<ebl, 256, 0, stream>>>(ei, a_src, a_dst, mx, denom, E, N);

  // 4) weighted aggregation, wave per edge
  long long nwaves = (long long)E + N;
  gat_edge_aggregate<<<(unsigned)((nwaves + 7) / 8), 256, 0, stream>>>(
      ei, a_src, a_dst, mx, denom, (const float4*)xw, out, E, N);
}